// Aggregation_76776835383365
// MI455X (gfx1250) — compile-verified
//
#include <hip/hip_runtime.h>
#include <hip/hip_bf16.h>

// Problem constants (match reference)
#define N_SENTS 64
#define N_ENTS  16
#define BATCH   64
#define DIM     512
#define HEADS   8
#define HD      64
#define BT      16              // batch tile per block
#define NBT     (BATCH / BT)    // 4
#define EBP     (BT + 1)        // LDS row pad to dodge bank conflicts

typedef float v2f __attribute__((ext_vector_type(2)));
typedef float v4f __attribute__((ext_vector_type(4)));
typedef float v8f __attribute__((ext_vector_type(8)));

// ---------------------------------------------------------------------------
// Kernel 1: fold the (single, broadcast) query through wq and wk.
//   q[r]    = bq[r] + sum_d query_param[d] * wq[r][d]
//   F[d][h] = sum_j wk[h*64+j][d] * q[h*64+j]
// The bk contribution to scores (q_h . bk_h) is constant over keys per head
// and cancels in the softmax, so bk is never needed.
// ---------------------------------------------------------------------------
__global__ __launch_bounds__(256)
void fold_kernel(const float* __restrict__ qp,
                 const float* __restrict__ wq,
                 const float* __restrict__ wk,
                 const float* __restrict__ bq,
                 float* __restrict__ F /* [DIM][HEADS] */) {
  __shared__ float sq[DIM];
  const int t = threadIdx.x;

  for (int r = t; r < DIM; r += 256) {
    float acc = bq[r];
    const float* wr = wq + (size_t)r * DIM;
    for (int d = 0; d < DIM; ++d) acc += qp[d] * wr[d];
    sq[r] = acc;
  }
  __syncthreads();

  for (int idx = t; idx < DIM * HEADS; idx += 256) {
    const int d = idx >> 3, h = idx & 7;
    float acc = 0.f;
    #pragma unroll 8
    for (int j = 0; j < HD; ++j)
      acc += wk[(size_t)(h * HD + j) * DIM + d] * sq[h * HD + j];
    F[(size_t)d * HEADS + h] = acc;
  }
}

// ---------------------------------------------------------------------------
// Kernel 2: per (sentence, batch-tile) fused attention.
//   scores -> softmax(16 ents) -> ebar (attention-weighted entity avg per head)
//   -> ctx^T = Wv_h @ ebar_h^T (+bv)   [v_wmma_f32_16x16x4_f32]
//   -> out^T = wo @ ctx^T (+bo)        [v_wmma_f32_16x16x4_f32]
// ---------------------------------------------------------------------------
__global__ __launch_bounds__(256)
void attn_kernel(const float* __restrict__ ents,
                 const float* __restrict__ Fg,
                 const float* __restrict__ wv,
                 const float* __restrict__ wo,
                 const float* __restrict__ bv,
                 const float* __restrict__ bo,
                 float* __restrict__ out) {
  extern __shared__ float lds[];
  float* sF    = lds;                               // DIM*HEADS            (16 KB)
  float* sAttn = sF + DIM * HEADS;                  // N_ENTS*BT*HEADS      ( 8 KB)
  float* sEbar = sAttn + N_ENTS * BT * HEADS;       // 2*DIM*EBP            (68 KB)
  float* sCtx  = sEbar + 2 * DIM * EBP;             // DIM*EBP              (34 KB)

  const int s    = blockIdx.x / NBT;
  const int bt   = blockIdx.x % NBT;
  const int tid  = threadIdx.x;
  const int lane = tid & 31;
  const int wave = tid >> 5;
  const int ln16 = lane & 15;
  const int hi16 = lane >> 4;     // 0 for lanes 0-15, 1 for lanes 16-31

  // Stage the folded query matrix.
  for (int i = tid; i < DIM * HEADS; i += 256) sF[i] = Fg[i];
  __syncthreads();

  // ---- Phase 1: scores[e][b][h] = (E(e,b,:) . F[:,h]) / 8 ----------------
  {
    const int e = tid >> 4, b = tid & 15;
    const v4f* ep = (const v4f*)(ents +
        ((size_t)(s * N_ENTS + e) * BATCH + bt * BT + b) * DIM);
    float acc[HEADS];
    #pragma unroll
    for (int h = 0; h < HEADS; ++h) acc[h] = 0.f;
    for (int d4 = 0; d4 < DIM / 4; ++d4) {
      v4f ev = ep[d4];
      const float* fr = sF + (size_t)(d4 * 4) * HEADS;
      #pragma unroll
      for (int h = 0; h < HEADS; ++h)
        acc[h] += ev.x * fr[h] + ev.y * fr[HEADS + h] +
                  ev.z * fr[2 * HEADS + h] + ev.w * fr[3 * HEADS + h];
    }
    #pragma unroll
    for (int h = 0; h < HEADS; ++h)
      sAttn[(e * BT + b) * HEADS + h] = acc[h] * 0.125f;   // 1/sqrt(64)
  }
  __syncthreads();

  // ---- Phase 2: softmax over the sentence's 16 entities ------------------
  if (tid < BT * HEADS) {
    const int b = tid >> 3, h = tid & 7;
    float m = -3.402823466e38f;
    #pragma unroll
    for (int e = 0; e < N_ENTS; ++e)
      m = fmaxf(m, sAttn[(e * BT + b) * HEADS + h]);
    float p[N_ENTS];
    float sum = 0.f;
    #pragma unroll
    for (int e = 0; e < N_ENTS; ++e) {
      p[e] = __expf(sAttn[(e * BT + b) * HEADS + h] - m);
      sum += p[e];
    }
    const float inv = 1.f / sum;
    #pragma unroll
    for (int e = 0; e < N_ENTS; ++e)
      sAttn[(e * BT + b) * HEADS + h] = p[e] * inv;
  }
  __syncthreads();

  // ---- Phase 3: per head-pair, ebar then ctx GEMM ------------------------
  for (int hp = 0; hp < HEADS / 2; ++hp) {
    // ebar_h[d][b] = sum_e attn[e][b][h] * E(e,b,d)   (re-read E from L2)
    {
      const int b     = tid & 15;
      const int hs    = (tid >> 4) & 1;          // which head of the pair
      const int h     = hp * 2 + hs;
      const int dbase = (tid >> 5) * 64;         // 64 d-values per thread
      v4f acc[16];
      #pragma unroll
      for (int i = 0; i < 16; ++i) acc[i] = (v4f){0.f, 0.f, 0.f, 0.f};
      for (int e = 0; e < N_ENTS; ++e) {
        const float w = sAttn[(e * BT + b) * HEADS + h];
        const v4f* ep = (const v4f*)(ents +
            ((size_t)(s * N_ENTS + e) * BATCH + bt * BT + b) * DIM + dbase);
        #pragma unroll
        for (int i = 0; i < 16; ++i) {
          v4f ev = ep[i];
          acc[i].x += w * ev.x; acc[i].y += w * ev.y;
          acc[i].z += w * ev.z; acc[i].w += w * ev.w;
        }
      }
      float* eb = sEbar + (size_t)hs * DIM * EBP;
      #pragma unroll
      for (int i = 0; i < 16; ++i) {
        const int d = dbase + i * 4;
        eb[(d    ) * EBP + b] = acc[i].x;
        eb[(d + 1) * EBP + b] = acc[i].y;
        eb[(d + 2) * EBP + b] = acc[i].z;
        eb[(d + 3) * EBP + b] = acc[i].w;
      }
    }
    __syncthreads();

    // ctx^T[m][n] = Wv_h @ ebar_h^T ; M=64 (2 heads x 4 tiles over 8 waves)
    {
      const int hs    = wave >> 2;               // 0..1 -> head of the pair
      const int h     = hp * 2 + hs;
      const int mtile = wave & 3;                // 0..3
      const float* eb = sEbar + (size_t)hs * DIM * EBP;
      const int row   = h * HD + mtile * 16 + ln16;   // row of wv
      const float* wr = wv + (size_t)row * DIM;
      v8f c = {};
      for (int k = 0; k < DIM; k += 4) {
        const int ke = k + hi16 * 2;
        // A (16x4, MxK): lane=M, VGPR0/1 = K {0,1} (lanes 0-15) / {2,3} (16-31)
        v2f a;  a.x = wr[ke];               a.y = wr[ke + 1];
        // B (4x16, KxN): mirrored split across lane halves
        v2f bm; bm.x = eb[ke * EBP + ln16]; bm.y = eb[(ke + 1) * EBP + ln16];
        c = __builtin_amdgcn_wmma_f32_16x16x4_f32(
                false, a, false, bm, (short)0, c, false, false);
      }
      #pragma unroll
      for (int r = 0; r < 8; ++r) {
        const int m = h * HD + mtile * 16 + r + hi16 * 8;  // C/D layout
        sCtx[m * EBP + ln16] = c[r] + bv[m];   // sum(attn)=1 -> bv passes thru
      }
    }
    __syncthreads();
  }

  // ---- Phase 4: out^T = wo @ ctx^T + bo ; M=512 -> 32 tiles over 8 waves --
  for (int t4 = 0; t4 < 4; ++t4) {
    const int mtile = wave * 4 + t4;             // 0..31
    const int row   = mtile * 16 + ln16;
    const float* wr = wo + (size_t)row * DIM;
    v8f c = {};
    for (int k = 0; k < DIM; k += 4) {
      const int ke = k + hi16 * 2;
      v2f a;  a.x = wr[ke];                 a.y = wr[ke + 1];
      v2f bm; bm.x = sCtx[ke * EBP + ln16]; bm.y = sCtx[(ke + 1) * EBP + ln16];
      c = __builtin_amdgcn_wmma_f32_16x16x4_f32(
              false, a, false, bm, (short)0, c, false, false);
    }
    #pragma unroll
    for (int r = 0; r < 8; ++r) {
      const int m = mtile * 16 + r + hi16 * 8;
      out[((size_t)s * BATCH + bt * BT + ln16) * DIM + m] = c[r] + bo[m];
    }
  }
}

// ---------------------------------------------------------------------------
extern "C" void kernel_launch(void* const* d_in, const int* in_sizes, int n_in,
                              void* d_out, int out_size, void* d_ws, size_t ws_size,
                              hipStream_t stream) {
  const float* ents = (const float*)d_in[0];
  const float* qp   = (const float*)d_in[1];
  const float* wq   = (const float*)d_in[2];
  const float* wk   = (const float*)d_in[3];
  const float* wv   = (const float*)d_in[4];
  const float* wo   = (const float*)d_in[5];
  const float* bq   = (const float*)d_in[6];
  // d_in[7] = bk : constant over keys per head -> cancels in softmax, unused
  const float* bv   = (const float*)d_in[8];
  const float* bo   = (const float*)d_in[9];
  // d_in[10] = n_sents (compile-time constant here)

  float* F   = (float*)d_ws;     // [DIM][HEADS] folded query, 16 KB
  float* out = (float*)d_out;    // [N_SENTS][BATCH][DIM]

  fold_kernel<<<1, 256, 0, stream>>>(qp, wq, wk, bq, F);

  const size_t lds_bytes =
      (size_t)(DIM * HEADS + N_ENTS * BT * HEADS + 2 * DIM * EBP + DIM * EBP) *
      sizeof(float);   // ~126 KB, well under the 320 KB WGP limit
  attn_kernel<<<N_SENTS * NBT, 256, lds_bytes, stream>>>(
      ents, F, wv, wo, bv, bo, out);
}